// WOS_55576876810250
// MI455X (gfx1250) — compile-verified
//
#include <hip/hip_runtime.h>
#include <stdint.h>

// Problem constants (match reference)
#define TKK 3
#define TCC 3
#define TNC 16
#define TD  27      // C*K*K
#define TMD 54      // 2*D
#define TB  4
#define THH 128
#define TWW 128
#define THREADS 256

typedef unsigned int u32;
typedef u32 u32x4 __attribute__((ext_vector_type(4)));
typedef int i32x4 __attribute__((ext_vector_type(4)));
typedef int i32x8 __attribute__((ext_vector_type(8)));

__global__ __launch_bounds__(THREADS)
void wos_kernel(const float* __restrict__ x,
                const float* __restrict__ mask,
                const float* __restrict__ weight,
                const float* __restrict__ bias,
                float* __restrict__ out) {
    __shared__ float  s_weight[TNC * TMD];     // 3456 B, filled by TDM
    __shared__ float2 s_wpair[TNC * TD];       // 3456 B, (w[d], w[d+27]) pairs
    __shared__ float  s_patch[THREADS * TD];   // 27648 B, per-thread patch copy

    // ---- TDM: DMA the (16 x 54) fp32 weight table Global -> LDS (wave 0) ----
    if (threadIdx.x < 32) {
        u32 lds_addr = (u32)(uintptr_t)&s_weight[0];          // addrspace(3) offset
        unsigned long long ga = (unsigned long long)(uintptr_t)weight;
        u32x4 g0;
        g0.x = 1u;                                            // count=1 (valid D#)
        g0.y = lds_addr;                                      // lds_addr
        g0.z = (u32)(ga & 0xffffffffull);                     // global_addr[31:0]
        g0.w = (u32)((ga >> 32) & 0x1ffffffull) | (2u << 30); // global_addr[56:32] | type=2
        i32x8 g1;
        g1[0] = (int)(2u << 16);            // data_size=2 -> 4 bytes; no mask/flags
        g1[1] = (int)(((u32)TMD) << 16);    // tensor_dim0[15:0] = 54
        g1[2] = (int)(((u32)TNC) << 16);    // tensor_dim0 hi=0, tensor_dim1[15:0]=16
        g1[3] = (int)(((u32)TMD) << 16);    // tensor_dim1 hi=0, tile_dim0=54
        g1[4] = (int)((u32)TNC);            // tile_dim1=16, tile_dim2=0
        g1[5] = (int)((u32)TMD);            // tensor_dim0_stride = 54
        g1[6] = 0;
        g1[7] = 0;
        i32x4 gz4 = {0, 0, 0, 0};
        i32x8 gz8 = {0, 0, 0, 0, 0, 0, 0, 0};
        __builtin_amdgcn_tensor_load_to_lds(g0, g1, gz4, gz4, gz8, 0);
        __builtin_amdgcn_s_wait_tensorcnt(0);
    }

    // ---- load 3x3x3 patch for this pixel (zero-padded 'same') ----
    int idx = blockIdx.x * THREADS + threadIdx.x;   // 0 .. B*H*W-1
    int px = idx & (TWW - 1);
    int py = (idx >> 7) & (THH - 1);
    int pb = idx >> 14;
    const float* xb = x + pb * (TCC * THH * TWW);

    float patch[TD];
#pragma unroll
    for (int cc = 0; cc < TCC; ++cc) {
#pragma unroll
        for (int dy = 0; dy < 3; ++dy) {
#pragma unroll
            for (int dx = 0; dx < 3; ++dx) {
                int yy = py + dy - 1;
                int xx = px + dx - 1;
                bool ok = (yy >= 0) & (yy < THH) & (xx >= 0) & (xx < TWW);
                float v = ok ? xb[cc * THH * TWW + yy * TWW + xx] : 0.0f;
                int d = cc * 9 + dy * 3 + dx;      // unfold ordering: c-major, i*k+j
                patch[d] = v;
                s_patch[threadIdx.x * TD + d] = v;
            }
        }
    }

    __syncthreads();   // TDM data visible to all waves

    // ---- restage weights as (w[c][d], w[c][d+27]) float2 pairs ----
    for (int t = threadIdx.x; t < TNC * TD; t += THREADS) {
        int c = t / TD;
        int d = t - c * TD;
        float2 pr;
        pr.x = s_weight[c * TMD + d];
        pr.y = s_weight[c * TMD + d + TD];
        s_wpair[t] = pr;
    }
    __syncthreads();

    // ---- per-channel weighted order statistic ----
#pragma unroll 1
    for (int c = 0; c < TNC; ++c) {
        float bcap = bias[c];                       // uniform -> scalar load

        // Build sort keys: magnitude bits (quantized) | sign<<5 | d.
        // Sorting 27 magnitudes descending determines the full 54-sort:
        // positions 0..26 are +|a| descending, 27..53 are -|a| mirrored.
        u32 key[32];
#pragma unroll
        for (int d = 0; d < TD; ++d) {
            float a = patch[d] + mask[c * TD + d];  // mask read is uniform -> s_load
            u32 mbits = __float_as_uint(fabsf(a));
            u32 meta = (u32)d | ((a >= 0.0f) ? 32u : 0u);
            key[d] = (mbits & ~63u) | meta;
        }
#pragma unroll
        for (int d = TD; d < 32; ++d) key[d] = 0u;  // sentinels sort to the end

        // Batcher odd-even mergesort, 32 keys, descending (191 comparators,
        // fully unrolled, registers only; CE = v_max_u32 + v_min_u32).
#pragma unroll
        for (int p = 1; p < 32; p <<= 1) {
#pragma unroll
            for (int k = p; k >= 1; k >>= 1) {
#pragma unroll
                for (int j = k & (p - 1); j + k < 32; j += 2 * k) {
#pragma unroll
                    for (int i = 0; i < k && i + j + k < 32; ++i) {
                        if (((i + j) / (p * 2)) == ((i + j + k) / (p * 2))) {
                            u32 a0 = key[i + j], b0 = key[i + j + k];
                            key[i + j]     = a0 > b0 ? a0 : b0;
                            key[i + j + k] = a0 > b0 ? b0 : a0;
                        }
                    }
                }
            }
        }

        // Weighted cumsum over the implicit 54-element sorted order.
        // Position k<27 holds element key[k] with value +|a|; position 27+t
        // holds element key[26-t] with value -|a|.  One ds_load_b64 per
        // element fetches both halves' weights.
        float acc = 0.0f;
        u32 selmeta = key[0] & 63u;   // li clipped to 0 if cumsum never <= bias
        u32 selneg = 0u;
        float wB[TD];
#pragma unroll
        for (int e = 0; e < TD; ++e) {
            u32 meta = key[e] & 63u;
            u32 d = meta & 31u;
            bool spos = (meta & 32u) != 0u;          // 1 iff a_d >= 0
            float2 pr = s_wpair[c * TD + (int)d];
            float wa = spos ? pr.x : pr.y;           // weight at first-half pos
            wB[e]    = spos ? pr.y : pr.x;           // weight at second-half pos
            acc += wa;
            bool le = acc <= bcap;
            selmeta = le ? meta : selmeta;
        }
#pragma unroll
        for (int t = 0; t < TD; ++t) {
            int e = TD - 1 - t;                      // reverse element order
            acc += wB[e];
            bool le = acc <= bcap;
            selmeta = le ? (key[e] & 63u) : selmeta;
            selneg  = le ? 1u : selneg;
        }

        // Recover the exact selected value (keys were magnitude-quantized).
        u32 dsel = selmeta & 31u;
        float a = s_patch[threadIdx.x * TD + dsel] + mask[c * TD + dsel];
        float v = fabsf(a);
        float yv = selneg ? -v : v;

        out[((pb * TNC + c) * THH + py) * TWW + px] = yv;
    }
}

extern "C" void kernel_launch(void* const* d_in, const int* in_sizes, int n_in,
                              void* d_out, int out_size, void* d_ws, size_t ws_size,
                              hipStream_t stream) {
    const float* x      = (const float*)d_in[0];
    const float* mask   = (const float*)d_in[1];
    const float* weight = (const float*)d_in[2];
    const float* bias   = (const float*)d_in[3];
    float* out = (float*)d_out;

    dim3 grid((TB * THH * TWW) / THREADS);   // 65536 pixels / 256 = 256 blocks
    wos_kernel<<<grid, THREADS, 0, stream>>>(x, mask, weight, bias, out);
}